// GCNLayer_9912784519225
// MI455X (gfx1250) — compile-verified
//
#include <hip/hip_runtime.h>
#include <hip/hip_bf16.h>

// ---------------------------------------------------------------------------
// GCN layer on gfx1250:
//   H = scatter-add over edges (SpMM)          -> f32 atomics into d_ws
//   out = relu(H @ W + b)                       -> V_WMMA_F32_16X16X4_F32
// ---------------------------------------------------------------------------

#define N_NODES 40000
#define N_EDGES 640000
#define IN_DIM  128
#define OUT_DIM 128

typedef __attribute__((ext_vector_type(2))) float v2f;
typedef __attribute__((ext_vector_type(8))) float v8f;

// ---------------------------------------------------------------- zero H ----
__global__ void gcn_zero_kernel(float4* __restrict__ p, int n4) {
    int i = blockIdx.x * blockDim.x + threadIdx.x;
    if (i < n4) p[i] = make_float4(0.f, 0.f, 0.f, 0.f);
}

// ------------------------------------------------------------- SpMM scatter -
// One wave32 per edge. Lane l handles floats [4l, 4l+4) of the 128-wide row:
// coalesced 512B read of X[src], 4 native f32 atomic adds into H[dst].
__global__ void gcn_scatter_kernel(const int* __restrict__ edge_index,
                                   const float* __restrict__ edge_weight,
                                   const float* __restrict__ X,
                                   float* __restrict__ H) {
    int gid  = blockIdx.x * blockDim.x + threadIdx.x;
    int e    = gid >> 5;           // wave32: one edge per wave
    int lane = gid & 31;
    if (e >= N_EDGES) return;

    int   src = edge_index[e];              // edge_index[0][e]
    int   dst = edge_index[N_EDGES + e];    // edge_index[1][e]
    float w   = edge_weight[e];

    const float4* xr = (const float4*)(X + (size_t)src * IN_DIM);
    float4 v = xr[lane];

    float* h = H + (size_t)dst * IN_DIM + lane * 4;
    unsafeAtomicAdd(h + 0, w * v.x);   // -> global_atomic_add_f32
    unsafeAtomicAdd(h + 1, w * v.y);
    unsafeAtomicAdd(h + 2, w * v.z);
    unsafeAtomicAdd(h + 3, w * v.w);
}

// ------------------------------------------- WMMA GEMM + bias + ReLU -------
// Block = 128 threads = 4 waves. Wave w computes rows [base, base+16) x all
// 128 output columns as eight 16x16 tiles. K is consumed 4 at a time with
// V_WMMA_F32_16X16X4_F32; the A fragment is loaded once per K-chunk and
// reused for all 8 column tiles (8 accumulators).
//
// ISA layouts (wave32):
//   A 16x4 f32 : lane<16 -> M=lane, regs {K=k0,k0+1}; lane>=16 -> {K=k0+2,k0+3}
//   B 4x16 f32 : lane<16 -> N=lane, regs {K=k0,k0+1}; lane>=16 -> {K=k0+2,k0+3}
//   D 16x16 f32: vgpr r  -> lanes 0-15: (M=r, N=lane); lanes 16-31: (M=r+8, N=lane-16)
__global__ void gcn_gemm_bias_relu_kernel(const float* __restrict__ H,
                                          const float* __restrict__ W,
                                          const float* __restrict__ bias,
                                          float* __restrict__ out) {
    int wave = threadIdx.x >> 5;
    int lane = threadIdx.x & 31;
    int rowBase = (blockIdx.x * 4 + wave) * 16;

    int mn  = lane & 15;        // M for A-frag, N for B-frag / D
    int kHi = lane >> 4;        // 0 or 1: K sub-pair select

    v8f acc[8];
#pragma unroll
    for (int t = 0; t < 8; ++t) acc[t] = v8f{};

    for (int k0 = 0; k0 < IN_DIM; k0 += 4) {
        // A fragment from H (row-major [N_NODES][IN_DIM])
        const float* ap = H + (size_t)(rowBase + mn) * IN_DIM + (k0 + 2 * kHi);
        v2f a;
        a.x = ap[0];
        a.y = ap[1];

#pragma unroll
        for (int t = 0; t < 8; ++t) {
            int colBase = t * 16;
            // B fragment from W (row-major [IN_DIM][OUT_DIM])
            const float* bp = W + (size_t)(k0 + 2 * kHi) * OUT_DIM + colBase + mn;
            v2f bf;
            bf.x = bp[0];
            bf.y = bp[OUT_DIM];
            acc[t] = __builtin_amdgcn_wmma_f32_16x16x4_f32(
                /*neg_a=*/false, a, /*neg_b=*/false, bf,
                /*c_mod=*/(short)0, acc[t],
                /*reuse_a=*/false, /*reuse_b=*/false);
        }
    }

    // Epilogue: bias + ReLU, write D per ISA C/D layout.
    int mOff = kHi * 8;
#pragma unroll
    for (int t = 0; t < 8; ++t) {
        int colBase = t * 16;
        float bb = bias[colBase + mn];
#pragma unroll
        for (int r = 0; r < 8; ++r) {
            float v = acc[t][r] + bb;
            out[(size_t)(rowBase + mOff + r) * OUT_DIM + colBase + mn] =
                v > 0.f ? v : 0.f;
        }
    }
}

// ---------------------------------------------------------------------------
extern "C" void kernel_launch(void* const* d_in, const int* in_sizes, int n_in,
                              void* d_out, int out_size, void* d_ws, size_t ws_size,
                              hipStream_t stream) {
    const int*   edge_index  = (const int*)d_in[0];   // [2, N_EDGES] int32
    const float* edge_weight = (const float*)d_in[1]; // [N_EDGES]
    const float* X           = (const float*)d_in[2]; // [N_NODES, IN_DIM]
    const float* W           = (const float*)d_in[3]; // [IN_DIM, OUT_DIM]
    const float* b           = (const float*)d_in[4]; // [OUT_DIM]
    float*       out         = (float*)d_out;         // [N_NODES, OUT_DIM]
    float*       H           = (float*)d_ws;          // scratch [N_NODES, IN_DIM]

    // 1) H = 0
    int n4 = N_NODES * IN_DIM / 4;
    gcn_zero_kernel<<<(n4 + 255) / 256, 256, 0, stream>>>((float4*)H, n4);

    // 2) H[dst] += w * X[src]  (wave32 per edge, 8 edges per 256-thread block)
    long long total = (long long)N_EDGES * 32;
    gcn_scatter_kernel<<<(int)((total + 255) / 256), 256, 0, stream>>>(
        edge_index, edge_weight, X, H);

    // 3) out = relu(H @ W + b): 2500 row tiles of 16, 4 waves per block
    gcn_gemm_bias_relu_kernel<<<2500 / 4, 128, 0, stream>>>(H, W, b, out);
}